// VotingModule_40613210751459
// MI455X (gfx1250) — compile-verified
//
#include <hip/hip_runtime.h>
#include <hip/hip_bf16.h>

#define NPTS   200000
#define C_IN   6
#define CDIM   256
#define KSEED  1024
#define NOUT   (3 + CDIM)      // 259
#define BN_EPS 1e-5f

typedef __attribute__((ext_vector_type(2))) float v2f;
typedef __attribute__((ext_vector_type(8))) float v8f;

// ---------------------------------------------------------------------------
// Kernel 1: farthest point sampling. One workgroup (1024 threads = 32 waves),
// 1023 dependent passes; min_d in global scratch (L2-resident), LDS argmax.
// ---------------------------------------------------------------------------
__global__ __launch_bounds__(1024) void fps_kernel(
    const float* __restrict__ points,   // [N,3]
    float* __restrict__ mind,           // [N] scratch
    int* __restrict__ idx_out)          // [K]
{
    __shared__ float sb[1024];
    __shared__ int   si[1024];

    const int tid = threadIdx.x;

    // init min distances
    for (int j = tid; j < NPTS; j += 1024) mind[j] = 1e10f;
    if (tid == 0) idx_out[0] = 0;
    __syncthreads();

    int last = 0;
    for (int i = 1; i < KSEED; ++i) {
        const float lx = points[last * 3 + 0];
        const float ly = points[last * 3 + 1];
        const float lz = points[last * 3 + 2];

        float best = -1.0f;
        int   bidx = 0x7fffffff;
        for (int j = tid; j < NPTS; j += 1024) {
            const float dx = points[j * 3 + 0] - lx;
            const float dy = points[j * 3 + 1] - ly;
            const float dz = points[j * 3 + 2] - lz;
            const float d  = dx * dx + dy * dy + dz * dz;
            float m = mind[j];
            m = fminf(m, d);
            mind[j] = m;
            if (m > best) { best = m; bidx = j; }   // keeps lowest j on ties
        }

        sb[tid] = best;
        si[tid] = bidx;
        __syncthreads();
        for (int s = 512; s > 0; s >>= 1) {
            if (tid < s) {
                const float ob = sb[tid + s];
                const int   oi = si[tid + s];
                if (ob > sb[tid] || (ob == sb[tid] && oi < si[tid])) {
                    sb[tid] = ob; si[tid] = oi;
                }
            }
            __syncthreads();
        }
        last = si[0];
        __syncthreads();
        if (tid == 0) idx_out[i] = last;
    }
}

// ---------------------------------------------------------------------------
// Kernel 2: gather seed points + seed_features = relu(features[idx] @ W_pfe)
// grid = K blocks, 256 threads (one output channel each). Only 1024 of the
// 200000 rows of the pointwise MLP are ever needed.
// ---------------------------------------------------------------------------
__global__ __launch_bounds__(256) void seed_pfe_kernel(
    const float* __restrict__ points,
    const float* __restrict__ features,
    const int* __restrict__ idx,
    const float* __restrict__ W_pfe,    // [6,256]
    float* __restrict__ seed_pts,       // [K,3]
    float* __restrict__ SF)             // [K,256]
{
    const int row = blockIdx.x;
    const int c   = threadIdx.x;
    const int id  = idx[row];
    if (c < 3) seed_pts[row * 3 + c] = points[id * 3 + c];
    float acc = 0.0f;
#pragma unroll
    for (int j = 0; j < C_IN; ++j)
        acc = fmaf(features[id * C_IN + j], W_pfe[j * CDIM + c], acc);
    SF[row * CDIM + c] = fmaxf(acc, 0.0f);
}

// ---------------------------------------------------------------------------
// Kernel 3: Hpre = SF @ W1  via V_WMMA_F32_16X16X4_F32.
// One wave per 16x16 output tile; 64 k-steps of K=4.
// A layout: lane%16 = M row; lanes<16 -> K=0,1  lanes>=16 -> K=2,3
// B layout: lane%16 = N col; lanes<16 -> K=0,1  lanes>=16 -> K=2,3
// C/D     : N = lane%16, M = vgpr + 8*(lane>=16)
// ---------------------------------------------------------------------------
__global__ __launch_bounds__(256) void gemm1_wmma(
    const float* __restrict__ SF,       // [K,256]
    const float* __restrict__ W1,       // [256,256]
    float* __restrict__ Hpre)           // [K,256]
{
    const int wave = threadIdx.x >> 5;
    const int lane = threadIdx.x & 31;
    const int tile = blockIdx.x * 8 + wave;     // 1024 tiles = 64 x 16
    const int tm   = tile >> 4;
    const int tn   = tile & 15;
    const int lr   = lane & 15;
    const int half = lane >> 4;

    const float* Arow = SF + (tm * 16 + lr) * CDIM + 2 * half;
    const float* Bcol = W1 + (2 * half) * CDIM + tn * 16 + lr;

    v8f acc = {};
    for (int k = 0; k < CDIM; k += 4) {
        __builtin_prefetch(Bcol + (k + 8) * CDIM, 0, 1);
        v2f a; a.x = Arow[k];          a.y = Arow[k + 1];
        v2f b; b.x = Bcol[k * CDIM];   b.y = Bcol[k * CDIM + CDIM];
        acc = __builtin_amdgcn_wmma_f32_16x16x4_f32(
            false, a, false, b, (short)0, acc, false, false);
    }

    const int m0  = tm * 16 + half * 8;
    const int col = tn * 16 + lr;
#pragma unroll
    for (int v = 0; v < 8; ++v)
        Hpre[(m0 + v) * CDIM + col] = acc[v];
}

// ---------------------------------------------------------------------------
// Kernel 4: BN batch stats over the K=1024 rows, folded into scale/shift.
// One block, 256 threads (one channel each); fully coalesced column sweeps.
// ---------------------------------------------------------------------------
__global__ __launch_bounds__(256) void bn_stats_kernel(
    const float* __restrict__ Hpre,
    const float* __restrict__ gamma,
    const float* __restrict__ beta,
    float* __restrict__ scale,
    float* __restrict__ shift)
{
    const int c = threadIdx.x;
    float s = 0.0f, ss = 0.0f;
    for (int r = 0; r < KSEED; ++r) {
        const float v = Hpre[r * CDIM + c];
        s += v; ss = fmaf(v, v, ss);
    }
    const float mean = s * (1.0f / KSEED);
    const float var  = ss * (1.0f / KSEED) - mean * mean;
    const float sc   = gamma[c] * rsqrtf(var + BN_EPS);
    scale[c] = sc;
    shift[c] = fmaf(-mean, sc, beta[c]);
}

// ---------------------------------------------------------------------------
// Kernel 5: residuals = relu(BN(Hpre)) @ W2 + b2, fused with the final adds.
// BN+ReLU applied on the fly when loading A. N=259 padded to 17 tiles of 16.
// ---------------------------------------------------------------------------
__global__ __launch_bounds__(256) void gemm2_wmma(
    const float* __restrict__ Hpre,     // [K,256]
    const float* __restrict__ scale,
    const float* __restrict__ shift,
    const float* __restrict__ W2,       // [256,259]
    const float* __restrict__ b2,       // [259]
    const float* __restrict__ seed_pts, // [K,3]
    const float* __restrict__ SF,       // [K,256]
    float* __restrict__ out)            // [K*3] ++ [K*256]
{
    const int wave = threadIdx.x >> 5;
    const int lane = threadIdx.x & 31;
    const int tile = blockIdx.x * 8 + wave;     // 1088 tiles = 64 x 17
    const int tm   = tile / 17;
    const int tn   = tile % 17;
    const int lr   = lane & 15;
    const int half = lane >> 4;

    const float* Arow = Hpre + (tm * 16 + lr) * CDIM + 2 * half;

    const int   col    = tn * 16 + lr;            // [0,272)
    const bool  bvalid = col < NOUT;
    const int   bcol   = bvalid ? col : 0;        // clamp, mask value instead
    const float bmask  = bvalid ? 1.0f : 0.0f;

    v8f acc = {};
    for (int k = 0; k < CDIM; k += 4) {
        const int k0 = k + 2 * half;
        v2f a;
        a.x = fmaxf(fmaf(Arow[k],     scale[k0],     shift[k0]),     0.0f);
        a.y = fmaxf(fmaf(Arow[k + 1], scale[k0 + 1], shift[k0 + 1]), 0.0f);
        v2f b;
        b.x = W2[k0 * NOUT + bcol] * bmask;
        b.y = W2[(k0 + 1) * NOUT + bcol] * bmask;
        acc = __builtin_amdgcn_wmma_f32_16x16x4_f32(
            false, a, false, b, (short)0, acc, false, false);
    }

    if (bvalid) {
        const float bias = b2[col];
#pragma unroll
        for (int v = 0; v < 8; ++v) {
            const int m = tm * 16 + half * 8 + v;
            const float r = acc[v] + bias;
            if (col < 3)
                out[m * 3 + col] = seed_pts[m * 3 + col] + r;
            else
                out[KSEED * 3 + m * CDIM + (col - 3)] =
                    SF[m * CDIM + (col - 3)] + r;
        }
    }
}

// ---------------------------------------------------------------------------
extern "C" void kernel_launch(void* const* d_in, const int* in_sizes, int n_in,
                              void* d_out, int out_size, void* d_ws, size_t ws_size,
                              hipStream_t stream)
{
    const float* points   = (const float*)d_in[0];
    const float* features = (const float*)d_in[1];
    const float* W_pfe    = (const float*)d_in[2];
    const float* W1       = (const float*)d_in[3];
    const float* gamma    = (const float*)d_in[4];
    const float* beta     = (const float*)d_in[5];
    const float* W2       = (const float*)d_in[6];
    const float* b2       = (const float*)d_in[7];
    (void)in_sizes; (void)n_in; (void)out_size; (void)ws_size;

    float* ws = (float*)d_ws;
    float* mind     = ws;                       // 200000
    int*   idx      = (int*)(ws + 200000);      // 1024
    float* seed_pts = ws + 201024;              // 3072
    float* SF       = ws + 204096;              // 262144
    float* Hpre     = ws + 466240;              // 262144
    float* scale    = ws + 728384;              // 256
    float* shift    = ws + 728640;              // 256

    float* out = (float*)d_out;

    fps_kernel     <<<1,    1024, 0, stream>>>(points, mind, idx);
    seed_pfe_kernel<<<KSEED, 256, 0, stream>>>(points, features, idx, W_pfe,
                                               seed_pts, SF);
    gemm1_wmma     <<<128,   256, 0, stream>>>(SF, W1, Hpre);
    bn_stats_kernel<<<1,     256, 0, stream>>>(Hpre, gamma, beta, scale, shift);
    gemm2_wmma     <<<136,   256, 0, stream>>>(Hpre, scale, shift, W2, b2,
                                               seed_pts, SF, out);
}